// SelfAttention_45457933861286
// MI455X (gfx1250) — compile-verified
//
#include <hip/hip_runtime.h>
#include <hip/hip_bf16.h>

#define EMBED 512
#define HEADS 8
#define SEQ   2048
#define HD    64
#define NB    2
#define NT    (SEQ / 16)   // 128 column tiles per row strip

typedef __attribute__((ext_vector_type(2))) float v2f;
typedef __attribute__((ext_vector_type(8))) float v8f;

__device__ __forceinline__ v2f ld2(const float* p) { return *(const v2f*)p; }

// ---------------------------------------------------------------------------
// Pair-transpose for B operands: dst holds K-pairs adjacent.
//   dst[((c/2)*R + r)*2 + (c&1)] = src[r*C + c]
// i.e. B[k][r] = src[r][k] stored so (k, k+1) for even k are a contiguous float2.
// ---------------------------------------------------------------------------
__global__ void mt_pair_transpose(float* __restrict__ dst, const float* __restrict__ src,
                                  int R, int C) {
    int idx = blockIdx.x * blockDim.x + threadIdx.x;
    if (idx < R * C) {
        int r = idx / C, c = idx % C;
        dst[((size_t)(c >> 1) * R + r) * 2 + (c & 1)] = src[(size_t)r * C + c];
    }
}

// ---------------------------------------------------------------------------
// Per-head projection: out[.,d] = sum_k in[.,k] * W[d][k] + bias[d]
// in:  [N,S,EMBED] viewed as [N,S,H,HD]
// Wt2: pair-interleaved B [HD/2][HD] float2
// mode 0 (Q): out [N,H,S,HD] plain
// mode 1 (K): out [N,H,HD/2,S] float2 pairs along d  (B layout for scores GEMM)
// mode 2 (V): out [N,H,S/2,HD] float2 pairs along s  (B layout for att@V GEMM)
// block = 128 (4 waves, wave w owns 16-col d-tile w); grid = N*H*(S/16)
// ---------------------------------------------------------------------------
__global__ void mt_proj(const float* __restrict__ in, const float* __restrict__ Wt2,
                        const float* __restrict__ bias, float* __restrict__ out,
                        int mode) {
    int blk  = blockIdx.x;
    int rt   = blk % NT;
    int nh   = blk / NT;
    int n    = nh / HEADS, h = nh % HEADS;
    int wave = threadIdx.x >> 5;
    int lane = threadIdx.x & 31;
    int mn   = lane & 15;       // m for A, n for B/C
    int kh   = lane >> 4;       // K half-select
    int d0   = wave * 16;

    const float* arow = in + (size_t)(n * SEQ + rt * 16 + mn) * EMBED + h * HD;

    v8f c = {};
    for (int t = 0; t < 16; ++t) {
        int k0 = 4 * t + 2 * kh;
        v2f a = ld2(arow + k0);
        v2f b = ld2(Wt2 + ((size_t)(k0 >> 1) * HD + d0 + mn) * 2);
        c = __builtin_amdgcn_wmma_f32_16x16x4_f32(false, a, false, b, (short)0, c,
                                                  false, false);
    }

    size_t hb = (size_t)nh;
    for (int r = 0; r < 8; ++r) {
        int row  = r + 8 * kh;
        int srow = rt * 16 + row;
        int d    = d0 + mn;
        float v  = c[r] + bias[d];
        if (mode == 1)        // K^T, d-paired: [(HD/2)][S] float2
            out[((hb * (HD / 2) + (d >> 1)) * SEQ + srow) * 2 + (d & 1)] = v;
        else if (mode == 2)   // V, s-paired: [(S/2)][HD] float2
            out[((hb * (SEQ / 2) + (srow >> 1)) * HD + d) * 2 + (srow & 1)] = v;
        else                  // Q plain [S][HD]
            out[(hb * SEQ + srow) * HD + d] = v;
    }
}

// ---------------------------------------------------------------------------
// Fused attention for one (n,h, 16 query rows):
//   scores -> causal mask -> softmax (row in LDS) -> += skewed QE -> att @ V
// block = 128 (4 waves); grid = N*H*(S/16); dynamic LDS = 16*S + 128 floats
// ---------------------------------------------------------------------------
__global__ void mt_attention(const float* __restrict__ Qp, const float* __restrict__ Ktr,
                             const float* __restrict__ Vp, const float* __restrict__ Et2,
                             float* __restrict__ z) {
    extern __shared__ float smem[];
    float* s_att = smem;               // [16][SEQ]
    float* s_red = smem + 16 * SEQ;    // [128]

    int blk  = blockIdx.x;
    int qt   = blk % NT;
    int nh   = blk / NT;
    int n    = nh / HEADS, h = nh % HEADS;
    int tid  = threadIdx.x;
    int wave = tid >> 5;
    int lane = tid & 31;
    int mn   = lane & 15;
    int kh   = lane >> 4;
    int m0   = qt * 16;

    const float* Qh = Qp  + (size_t)nh * SEQ * HD;
    const float* Kh = Ktr + (size_t)nh * HD * SEQ;   // pair-interleaved [(HD/2)][S]x2
    const float* Vh = Vp  + (size_t)nh * SEQ * HD;   // pair-interleaved [(S/2)][HD]x2

    // Preload the Q-tile A-fragments for all 16 K-steps (32 VGPRs)
    v2f aq[16];
    {
        const float* qrow = Qh + (size_t)(m0 + mn) * HD;
        for (int t = 0; t < 16; ++t) aq[t] = ld2(qrow + 4 * t + 2 * kh);
    }

    // ---- Phase 1: scores = Q K^T, causal mask (keep strict upper triangle) ----
    const float rscale = 0.044194173824159216f;   // 1/sqrt(512)
    for (int kt = wave; kt < NT; kt += 4) {
        int c0 = kt * 16;
        if (kt + 4 < NT)
            __builtin_prefetch(Kh + ((size_t)kh * SEQ + (kt + 4) * 16 + mn) * 2, 0, 3);
        v8f c = {};
        for (int t = 0; t < 16; ++t) {
            int kp = 2 * t + kh;   // (k0>>1), k0 = 4t+2kh
            v2f b = ld2(Kh + ((size_t)kp * SEQ + c0 + mn) * 2);
            c = __builtin_amdgcn_wmma_f32_16x16x4_f32(false, aq[t], false, b, (short)0, c,
                                                      false, false);
        }
        for (int r = 0; r < 8; ++r) {
            int row = r + 8 * kh;
            int gq  = m0 + row;
            int gc  = c0 + mn;
            float x = (gc > gq) ? c[r] * rscale : -1e20f * rscale;
            s_att[row * SEQ + gc] = x;
        }
    }
    __syncthreads();

    // ---- Phase 2: row softmax (8 threads per row) ----
    {
        int row = tid >> 3;
        int sub = tid & 7;
        float lmax = -3.4e38f;
        for (int c = sub; c < SEQ; c += 8) lmax = fmaxf(lmax, s_att[row * SEQ + c]);
        s_red[tid] = lmax;
        __syncthreads();
        float rmax = -3.4e38f;
        for (int i = 0; i < 8; ++i) rmax = fmaxf(rmax, s_red[row * 8 + i]);
        __syncthreads();
        float lsum = 0.f;
        for (int c = sub; c < SEQ; c += 8) {
            float e = __expf(s_att[row * SEQ + c] - rmax);
            s_att[row * SEQ + c] = e;
            lsum += e;
        }
        s_red[tid] = lsum;
        __syncthreads();
        float rsum = 0.f;
        for (int i = 0; i < 8; ++i) rsum += s_red[row * 8 + i];
        float inv = 1.0f / rsum;
        for (int c = sub; c < SEQ; c += 8) s_att[row * SEQ + c] *= inv;
    }
    __syncthreads();

    // ---- Phase 3: att[i][j] += Srel[i][j] = QE[i][S-1-i+j] (j <= i) ----
    for (int lt = wave; lt < NT; lt += 4) {
        int l0 = lt * 16;
        v8f c = {};
        for (int t = 0; t < 16; ++t) {
            int kp = 2 * t + kh;
            v2f b = ld2(Et2 + ((size_t)kp * SEQ + l0 + mn) * 2);
            c = __builtin_amdgcn_wmma_f32_16x16x4_f32(false, aq[t], false, b, (short)0, c,
                                                      false, false);
        }
        for (int r = 0; r < 8; ++r) {
            int row = r + 8 * kh;
            int gq  = m0 + row;
            int l   = l0 + mn;
            int j   = l - (SEQ - 1) + gq;     // j <= gq always (l <= S-1)
            if (j >= 0) s_att[row * SEQ + j] += c[r];
        }
    }
    __syncthreads();

    // ---- Phase 4: z_tile = att (16 x S) @ V (S x 64); wave w -> cols w*16.. ----
    int d0 = wave * 16;
    v8f acc = {};
    for (int kk = 0; kk < SEQ / 4; ++kk) {
        int k0 = 4 * kk + 2 * kh;
        v2f a = ld2(&s_att[mn * SEQ + k0]);                        // ds_load_b64
        v2f b = ld2(Vh + ((size_t)(k0 >> 1) * HD + d0 + mn) * 2);  // global_load_b64
        acc = __builtin_amdgcn_wmma_f32_16x16x4_f32(false, a, false, b, (short)0, acc,
                                                    false, false);
    }
    for (int r = 0; r < 8; ++r) {
        int row = r + 8 * kh;
        int s   = m0 + row;
        z[(size_t)(n * SEQ + s) * EMBED + h * HD + d0 + mn] = acc[r];
    }
}

// ---------------------------------------------------------------------------
// out = z [N*S, EMBED] @ Wo^T + bo, Wo^T pair-interleaved [(EMBED/2)][EMBED]x2
// grid = (N*S/16, EMBED/64), block 128 (wave w -> 16-col tile w)
// ---------------------------------------------------------------------------
__global__ void mt_out(const float* __restrict__ zbuf, const float* __restrict__ Wot2,
                       const float* __restrict__ bo, float* __restrict__ outp) {
    int wave = threadIdx.x >> 5;
    int lane = threadIdx.x & 31;
    int mn   = lane & 15;
    int kh   = lane >> 4;
    int m0   = blockIdx.x * 16;
    int col0 = blockIdx.y * 64 + wave * 16;

    const float* arow = zbuf + (size_t)(m0 + mn) * EMBED;
    v8f c = {};
    for (int kk = 0; kk < EMBED / 4; ++kk) {
        int k0 = 4 * kk + 2 * kh;
        v2f a = ld2(arow + k0);
        v2f b = ld2(Wot2 + ((size_t)(k0 >> 1) * EMBED + col0 + mn) * 2);
        c = __builtin_amdgcn_wmma_f32_16x16x4_f32(false, a, false, b, (short)0, c,
                                                  false, false);
    }
    for (int r = 0; r < 8; ++r) {
        int row = r + 8 * kh;
        outp[(size_t)(m0 + row) * EMBED + col0 + mn] = c[r] + bo[col0 + mn];
    }
}

// ---------------------------------------------------------------------------
extern "C" void kernel_launch(void* const* d_in, const int* in_sizes, int n_in,
                              void* d_out, int out_size, void* d_ws, size_t ws_size,
                              hipStream_t stream) {
    const float* v_in = (const float*)d_in[0];
    const float* k_in = (const float*)d_in[1];
    const float* q_in = (const float*)d_in[2];
    const float* Wv   = (const float*)d_in[3];
    const float* bv   = (const float*)d_in[4];
    const float* Wk   = (const float*)d_in[5];
    const float* bk   = (const float*)d_in[6];
    const float* Wq   = (const float*)d_in[7];
    const float* bq   = (const float*)d_in[8];
    const float* E    = (const float*)d_in[9];
    const float* Wo   = (const float*)d_in[10];
    const float* bo   = (const float*)d_in[11];
    float* outp = (float*)d_out;

    const size_t NHSD = (size_t)NB * SEQ * EMBED;   // 2,097,152 floats
    float* w   = (float*)d_ws;
    float* Qp  = w;                   // [N,H,S,HD]
    float* Ktr = Qp  + NHSD;          // [N,H,HD/2,S] float2 (d-paired)
    float* Vp  = Ktr + NHSD;          // [N,H,S/2,HD] float2 (s-paired)
    float* zb  = Vp  + NHSD;          // [N,S,EMBED]
    float* WqT = zb  + NHSD;          // pair-interleaved 64x64
    float* WkT = WqT + HD * HD;
    float* WvT = WkT + HD * HD;
    float* WoT = WvT + HD * HD;       // pair-interleaved [(EMBED/2)][EMBED]x2
    float* Et  = WoT + (size_t)EMBED * EMBED;   // pair-interleaved [(HD/2)][SEQ]x2

    // B-operand layout prep (pair-interleaved transposes)
    mt_pair_transpose<<<(HD * HD + 255) / 256, 256, 0, stream>>>(WvT, Wv, HD, HD);
    mt_pair_transpose<<<(HD * HD + 255) / 256, 256, 0, stream>>>(WkT, Wk, HD, HD);
    mt_pair_transpose<<<(HD * HD + 255) / 256, 256, 0, stream>>>(WqT, Wq, HD, HD);
    mt_pair_transpose<<<(EMBED * EMBED + 255) / 256, 256, 0, stream>>>(WoT, Wo, EMBED, EMBED);
    mt_pair_transpose<<<(SEQ * HD + 255) / 256, 256, 0, stream>>>(Et, E, SEQ, HD);

    // Projections (WMMA)
    int pgrid = NB * HEADS * NT;
    mt_proj<<<pgrid, 128, 0, stream>>>(q_in, WqT, bq, Qp, 0);
    mt_proj<<<pgrid, 128, 0, stream>>>(k_in, WkT, bk, Ktr, 1);
    mt_proj<<<pgrid, 128, 0, stream>>>(v_in, WvT, bv, Vp, 2);

    // Fused attention (WMMA + LDS softmax + skewed relative-position add)
    size_t smem = (16 * SEQ + 128) * sizeof(float);   // ~131.5 KB of the 320 KB WGP LDS
    mt_attention<<<pgrid, 128, smem, stream>>>(Qp, Ktr, Vp, Et, zb);

    // Output projection (WMMA)
    mt_out<<<dim3((NB * SEQ) / 16, EMBED / 64), 128, 0, stream>>>(zb, WoT, bo, outp);
}